// Conv2DUF_67654324847044
// MI455X (gfx1250) — compile-verified
//
#include <hip/hip_runtime.h>

// Conv2D 3x3 same-pad, C_in=C_out=16, 1024x1024, fp32, as implicit GEMM
// using V_WMMA_F32_16X16X4_F32 (CDNA5 fp32 matrix path, wave32).
// LDS input tile is channel-pair interleaved so each WMMA B fragment is a
// single aligned ds_load_b64.

typedef __attribute__((ext_vector_type(2))) float v2f;
typedef __attribute__((ext_vector_type(8))) float v8f;

constexpr int CIN  = 16;
constexpr int COUT = 16;
constexpr int HH   = 1024;
constexpr int WW   = 1024;

constexpr int TILE_ROWS = 8;            // output rows per block (one per wave)
constexpr int TILE_COLS = 64;           // output cols per block (4 WMMA N-tiles/wave)
constexpr int LDS_R  = TILE_ROWS + 2;   // 10 input rows incl. halo
constexpr int LDS_C  = TILE_COLS + 2;   // 66 input cols incl. halo
constexpr int LDS_CP = 68;              // padded col stride (in float-PAIRS)
constexpr int NKC    = 36;              // K = 144 = 36 chunks of 4

// xs layout: [cipair 0..7][r 0..9][c 0..67][ci&1] -> pair-interleaved channels
__global__ __launch_bounds__(256)
void conv3x3_wmma_f32(const float* __restrict__ x,
                      const float* __restrict__ w,
                      const float* __restrict__ bias,
                      float* __restrict__ out) {
  __shared__ float xs[(CIN / 2) * LDS_R * LDS_CP * 2];  // 43,520 B
  __shared__ float ws[NKC * 64];                        //  9,216 B

  const int tid     = threadIdx.x;
  const int tile_w0 = blockIdx.x * TILE_COLS;
  const int tile_h0 = blockIdx.y * TILE_ROWS;

  // ---- Stage weights into LDS in WMMA A-fragment order ----
  // k ordering: k = tap*16 + ci, tap = kh*3 + kw.
  // ws[kc*64 + lane*2 + j] = W[m = lane%16][k = 4*kc + j + 2*(lane/16)]
  for (int i = tid; i < NKC * 64; i += 256) {
    int kc     = i >> 6;
    int rem    = i & 63;
    int lane_s = rem >> 1;
    int j      = rem & 1;
    int hb     = lane_s >> 4;
    int m      = lane_s & 15;
    int k      = 4 * kc + j + 2 * hb;
    int tap    = k >> 4;
    int ci     = k & 15;
    int kh     = tap / 3;
    int kw     = tap - 3 * kh;
    ws[i] = w[((m * CIN + ci) * 3 + kh) * 3 + kw];
  }

  // ---- Stage input tile (+halo) into LDS (pair-interleaved), zero-fill OOB ----
  __builtin_prefetch(x + ((size_t)tile_h0) * WW + tile_w0, 0, 1); // global_prefetch_b8
  constexpr int NELT = CIN * LDS_R * LDS_C;
  for (int i = tid; i < NELT; i += 256) {
    int ci  = i / (LDS_R * LDS_C);
    int rem = i - ci * (LDS_R * LDS_C);
    int r   = rem / LDS_C;
    int c   = rem - r * LDS_C;
    int gr  = tile_h0 - 1 + r;
    int gc  = tile_w0 - 1 + c;
    float v = 0.0f;
    if ((unsigned)gr < (unsigned)HH && (unsigned)gc < (unsigned)WW)
      v = x[((size_t)ci * HH + gr) * WW + gc];
    xs[((((ci >> 1) * LDS_R + r) * LDS_CP + c) << 1) | (ci & 1)] = v;
  }
  __syncthreads();

  const int wv   = tid >> 5;      // wave id = output row within tile
  const int lane = tid & 31;
  const int n    = lane & 15;     // N (pixel) index within 16-wide tile
  const int hb   = lane >> 4;     // lane half

  // ---- Accumulators initialized with bias (C/D layout: M = v + 8*hb, N = n) ----
  v8f acc0, acc1, acc2, acc3;
  #pragma unroll
  for (int v = 0; v < 8; ++v) {
    float bv = bias[v + 8 * hb];
    acc0[v] = bv; acc1[v] = bv; acc2[v] = bv; acc3[v] = bv;
  }

  // ---- K loop: 36 chunks of 4; each chunk stays within one 3x3 tap ----
  #pragma unroll 4
  for (int kc = 0; kc < NKC; ++kc) {
    v2f a = *(const v2f*)(&ws[kc * 64 + lane * 2]);

    int tap = kc >> 2;
    int kh  = tap / 3;
    int kw  = tap - 3 * kh;
    int cp  = ((kc & 3) << 1) | hb;   // ci0/2; frag k = j + 2*hb -> floats (ci0, ci0+1)

    // One aligned b64 load per B fragment; nt tiles are +128B immediates.
    const v2f* pb =
        (const v2f*)&xs[(((cp * LDS_R) + (wv + kh)) * LDS_CP + n + kw) << 1];

    v2f b0 = pb[0];
    acc0 = __builtin_amdgcn_wmma_f32_16x16x4_f32(false, a, false, b0,
                                                 (short)0, acc0, false, false);
    v2f b1 = pb[16];
    acc1 = __builtin_amdgcn_wmma_f32_16x16x4_f32(false, a, false, b1,
                                                 (short)0, acc1, false, false);
    v2f b2 = pb[32];
    acc2 = __builtin_amdgcn_wmma_f32_16x16x4_f32(false, a, false, b2,
                                                 (short)0, acc2, false, false);
    v2f b3 = pb[48];
    acc3 = __builtin_amdgcn_wmma_f32_16x16x4_f32(false, a, false, b3,
                                                 (short)0, acc3, false, false);
  }

  // ---- Store: lane writes channel co = v + 8*hb, pixel tile_w0 + nt*16 + n ----
  const int h = tile_h0 + wv;
  #pragma unroll
  for (int v = 0; v < 8; ++v) {
    int co = v + 8 * hb;
    float* op = out + ((size_t)co * HH + h) * WW + tile_w0 + n;
    op[0]  = acc0[v];
    op[16] = acc1[v];
    op[32] = acc2[v];
    op[48] = acc3[v];
  }
}

extern "C" void kernel_launch(void* const* d_in, const int* in_sizes, int n_in,
                              void* d_out, int out_size, void* d_ws, size_t ws_size,
                              hipStream_t stream) {
  const float* x    = (const float*)d_in[0];
  const float* wgt  = (const float*)d_in[1];
  const float* bias = (const float*)d_in[2];
  float* out        = (float*)d_out;

  dim3 grid(WW / TILE_COLS, HH / TILE_ROWS);  // 16 x 128
  conv3x3_wmma_f32<<<grid, dim3(256), 0, stream>>>(x, wgt, bias, out);
}